// MLP_41059887349793
// MI455X (gfx1250) — compile-verified
//
#include <hip/hip_runtime.h>
#include <hip/hip_bf16.h>

typedef __attribute__((ext_vector_type(2))) float v2f;
typedef __attribute__((ext_vector_type(8))) float v8f;

// Edge-MLP: logits[m] = relu(concat(emb[src],emb[dst]) @ W1 + b1) @ W2 + b2
// GEMM [E,256]x[256,32] via V_WMMA_F32_16X16X4_F32.
// Each wave processes TWO 16-edge M-tiles sharing the same B (W1) operands,
// so each ds_load feeds 2 WMMAs.
//
// Output layout (flat, reference tuple order):
//   d_out[0   ..  E) : edge_logits (f32)
//   d_out[E   .. 2E) : edge_index[0] = src (int32 bit pattern)
//   d_out[2E  .. 3E) : edge_index[1] = dst (int32 bit pattern)

__global__ __launch_bounds__(256) void
edge_mlp_wmma_kernel(const float* __restrict__ node_emb,
                     const long long* __restrict__ edge,
                     const float* __restrict__ W1,   // [256,32] row-major
                     const float* __restrict__ b1,   // [32]
                     const float* __restrict__ W2,   // [32]
                     const float* __restrict__ b2,   // [1]
                     float* __restrict__ out,
                     int E)
{
    // B staging: for k-step kk (K=4kk..4kk+3), ntile n, lane l holds
    // float2 {W1[4kk+2*(l>>4)+0][N], W1[4kk+2*(l>>4)+1][N]}, N = n*16 + (l&15).
    // Matches ISA B layout: row K = vgpr + 2*(lane/16), col N = lane%16.
    __shared__ v2f ldsB[64 * 64];   // 32 KB

    const int t = threadIdx.x;
    for (int i = t; i < 64 * 64; i += 256) {
        int kk    = i >> 6;
        int rem   = i & 63;
        int ntile = rem >> 5;
        int l     = rem & 31;
        int hlw   = l >> 4;
        int n     = ntile * 16 + (l & 15);
        int k0    = 4 * kk + 2 * hlw;
        v2f w;
        w.x = W1[(k0 + 0) * 32 + n];
        w.y = W1[(k0 + 1) * 32 + n];
        ldsB[i] = w;
    }
    __syncthreads();

    const int wave = t >> 5;
    const int lane = t & 31;
    const int hl   = lane >> 4;   // half-wave select
    const int ml   = lane & 15;

    // Two 16-edge tiles per wave.
    const int base0 = (blockIdx.x * 8 + wave) * 32;
    const int base1 = base0 + 16;

    const int m0 = base0 + ml;
    const int m1 = base1 + ml;
    const bool v0 = (m0 < E);
    const bool v1 = (m1 < E);
    const int mc0 = v0 ? m0 : (E - 1);   // clamp: keep EXEC all-1s around WMMA
    const int mc1 = v1 ? m1 : (E - 1);

    const int s0 = (int)edge[2 * (long long)mc0 + 0];
    const int d0 = (int)edge[2 * (long long)mc0 + 1];
    const int s1 = (int)edge[2 * (long long)mc1 + 0];
    const int d1 = (int)edge[2 * (long long)mc1 + 1];

    // A layout: lane l supplies row M=l%16, K = 2*(l/16)+{0,1} -> one aligned b64/k-step.
    const float* pa0 = node_emb + (size_t)s0 * 128 + 2 * hl;   // tile0 dims   0..127
    const float* pb0 = node_emb + (size_t)d0 * 128 + 2 * hl;   // tile0 dims 128..255
    const float* pa1 = node_emb + (size_t)s1 * 128 + 2 * hl;   // tile1 dims   0..127
    const float* pb1 = node_emb + (size_t)d1 * 128 + 2 * hl;   // tile1 dims 128..255

    v8f c00 = {};   // tile0, H cols  0..15
    v8f c01 = {};   // tile0, H cols 16..31
    v8f c10 = {};   // tile1, H cols  0..15
    v8f c11 = {};   // tile1, H cols 16..31

    #pragma unroll 4
    for (int kk = 0; kk < 32; ++kk) {
        v2f a0  = *(const v2f*)(pa0 + 4 * kk);
        v2f a1  = *(const v2f*)(pa1 + 4 * kk);
        v2f bb0 = ldsB[kk * 64 + lane];
        v2f bb1 = ldsB[kk * 64 + 32 + lane];
        c00 = __builtin_amdgcn_wmma_f32_16x16x4_f32(false, a0, false, bb0, (short)0, c00, false, false);
        c01 = __builtin_amdgcn_wmma_f32_16x16x4_f32(false, a0, false, bb1, (short)0, c01, false, false);
        c10 = __builtin_amdgcn_wmma_f32_16x16x4_f32(false, a1, false, bb0, (short)0, c10, false, false);
        c11 = __builtin_amdgcn_wmma_f32_16x16x4_f32(false, a1, false, bb1, (short)0, c11, false, false);
    }
    #pragma unroll 4
    for (int kk = 32; kk < 64; ++kk) {
        v2f a0  = *(const v2f*)(pb0 + 4 * (kk - 32));
        v2f a1  = *(const v2f*)(pb1 + 4 * (kk - 32));
        v2f bb0 = ldsB[kk * 64 + lane];
        v2f bb1 = ldsB[kk * 64 + 32 + lane];
        c00 = __builtin_amdgcn_wmma_f32_16x16x4_f32(false, a0, false, bb0, (short)0, c00, false, false);
        c01 = __builtin_amdgcn_wmma_f32_16x16x4_f32(false, a0, false, bb1, (short)0, c01, false, false);
        c10 = __builtin_amdgcn_wmma_f32_16x16x4_f32(false, a1, false, bb0, (short)0, c10, false, false);
        c11 = __builtin_amdgcn_wmma_f32_16x16x4_f32(false, a1, false, bb1, (short)0, c11, false, false);
    }

    // Epilogue. C layout: VGPR r, lane l -> edge M = r + 8*(l/16), H col N = l%16 (+16 for c*1).
    const float b1a = b1[ml];
    const float b1b = b1[16 + ml];
    const float w2a = W2[ml];
    const float w2b = W2[16 + ml];
    const float bias2 = b2[0];

    // Fused edge_index output (src/dst already in registers on lanes 0..15).
    int* iout = (int*)out;
    if (hl == 0) {
        if (v0) { iout[E + m0] = s0; iout[2 * E + m0] = d0; }
        if (v1) { iout[E + m1] = s1; iout[2 * E + m1] = d1; }
    }

    #pragma unroll
    for (int r = 0; r < 8; ++r) {
        // tile 0
        float h0 = c00[r] + b1a; h0 = h0 > 0.0f ? h0 : 0.0f;
        float h1 = c01[r] + b1b; h1 = h1 > 0.0f ? h1 : 0.0f;
        float p0 = h0 * w2a + h1 * w2b;
        p0 += __shfl_xor(p0, 8);
        p0 += __shfl_xor(p0, 4);
        p0 += __shfl_xor(p0, 2);
        p0 += __shfl_xor(p0, 1);
        // tile 1
        float g0 = c10[r] + b1a; g0 = g0 > 0.0f ? g0 : 0.0f;
        float g1 = c11[r] + b1b; g1 = g1 > 0.0f ? g1 : 0.0f;
        float p1 = g0 * w2a + g1 * w2b;
        p1 += __shfl_xor(p1, 8);
        p1 += __shfl_xor(p1, 4);
        p1 += __shfl_xor(p1, 2);
        p1 += __shfl_xor(p1, 1);

        const int me0 = base0 + r + 8 * hl;
        const int me1 = base1 + r + 8 * hl;
        if (ml == 0) {
            if (me0 < E) out[me0] = p0 + bias2;
            if (me1 < E) out[me1] = p1 + bias2;
        }
    }
}

extern "C" void kernel_launch(void* const* d_in, const int* in_sizes, int n_in,
                              void* d_out, int out_size, void* d_ws, size_t ws_size,
                              hipStream_t stream) {
    const float*     node_emb = (const float*)d_in[0];
    // d_in[1] = node_num (unused)
    const long long* edge     = (const long long*)d_in[2];
    const float*     W1       = (const float*)d_in[3];
    const float*     b1       = (const float*)d_in[4];
    const float*     W2       = (const float*)d_in[5];
    const float*     b2       = (const float*)d_in[6];
    float*           out      = (float*)d_out;

    const int E = in_sizes[2] / 2;          // edge is [E,2] int64
    const int edgesPerBlock = 8 * 32;       // 8 waves x two 16-edge tiles
    const int nblocks = (E + edgesPerBlock - 1) / edgesPerBlock;

    edge_mlp_wmma_kernel<<<nblocks, 256, 0, stream>>>(node_emb, edge, W1, b1, W2, b2, out, E);
}